// STHD_SpGAT_75814762709175
// MI455X (gfx1250) — compile-verified
//
#include <hip/hip_runtime.h>
#include <hip/hip_bf16.h>

typedef __attribute__((ext_vector_type(2))) float v2f;
typedef __attribute__((ext_vector_type(8))) float v8f;
typedef __attribute__((ext_vector_type(4))) unsigned int u32x4;
typedef __attribute__((ext_vector_type(4))) int i32x4;
typedef __attribute__((ext_vector_type(8))) int i32x8;

namespace {
constexpr int   NCELL = 20000;
constexpr int   K     = 20;
constexpr int   G     = 256;
constexpr int   H     = 8;
constexpr int   E     = 320000;
constexpr int   NT    = NCELL / 16;     // 1250 row tiles of 16 cells
constexpr int   TPB   = 8;              // tiles (waves) per block
constexpr int   ROWS  = TPB * 16;       // 128 rows staged per block
constexpr int   LDSPITCH = G + 4;       // 256 + 4 DWORD pad -> bank-conflict-free
constexpr int   CH    = G / 4;          // 64 k-chunks of 4
constexpr int   PANEL = CH * 64;        // 4096 floats per swizzled B panel
constexpr float NEG_SLOPE = 0.2f;

// workspace layout (floats)
constexpr size_t OFF_BSW  = 0;                              // 5 swizzled panels [5*PANEL]
constexpr size_t OFF_CC   = OFF_BSW  + (size_t)5 * PANEL;   // C[k] [32]
constexpr size_t OFF_LOGP = OFF_CC   + 32;                  // log(P+1e-8) [N*K]
constexpr size_t OFF_XLR  = OFF_LOGP + (size_t)NCELL * K;   // xl||xr [N*16]
constexpr size_t OFF_EEX  = OFF_XLR  + (size_t)NCELL * 16;  // per-edge e then exp [E]
constexpr size_t OFF_MKEY = OFF_EEX  + (size_t)E;           // segment max keys [N] (int)
constexpr size_t OFF_SSUM = OFF_MKEY + (size_t)NCELL;       // segment exp sums [N]
constexpr size_t OFF_ACC  = OFF_SSUM + (size_t)NCELL;       // [0]=ll_sum [1]=ce_sum
} // namespace

// ---------- helpers ----------
__device__ __forceinline__ int fkey(float f) {          // order-preserving float->int
  int i = __float_as_int(f);
  return (i >= 0) ? i : (i ^ 0x7FFFFFFF);
}
__device__ __forceinline__ float fdec(int k) {
  int i = (k >= 0) ? k : (k ^ 0x7FFFFFFF);
  return __int_as_float(i);
}
__device__ __forceinline__ float block_reduce_add(float v) {
  __shared__ float sm[8];
  for (int off = 16; off; off >>= 1) v += __shfl_down(v, off, 32);
  int lane = threadIdx.x & 31, w = threadIdx.x >> 5;
  if (lane == 0) sm[w] = v;
  __syncthreads();
  if (w == 0) {
    v = (lane < (int)(blockDim.x >> 5)) ? sm[lane] : 0.f;
    for (int off = 4; off; off >>= 1) v += __shfl_down(v, off, 32);
  }
  return v;   // valid on thread 0
}

__device__ __forceinline__ void tdm_load_rows(const float* gsrc, unsigned ldsByteOff,
                                              int remRows) {
  // D# group 0: count=1 | lds_addr | global_addr(57b) | type=2
  unsigned long long ga = (unsigned long long)(const void*)gsrc;
  u32x4 g0;
  g0.x = 1u;                                   // count = 1 valid descriptor
  g0.y = ldsByteOff;                           // lds_addr (bytes)
  g0.z = (unsigned)ga;                         // global_addr[31:0]
  g0.w = (unsigned)(ga >> 32) | 0x80000000u;   // global_addr[56:32] | type=2<<30
  // D# group 1
  i32x8 g1;
  g1[0] = (int)((2u << 16)      // data_size = 4 bytes
              | (1u << 20)      // pad_enable
              | (7u << 22)      // pad_interval: 256 DWORDs (one 1KB row)
              | (3u << 25));    // pad_amount: 4 DWORDs (16B) -> pitch 260 floats
  g1[1] = (int)((unsigned)(G & 0xffff) << 16);          // tensor_dim0 lo16 (=256)
  g1[2] = (int)(((unsigned)G >> 16) |
                ((unsigned)(remRows & 0xffff) << 16));  // dim0 hi16 | tensor_dim1 lo16
  g1[3] = (int)((unsigned)(G & 0xffff) << 16);          // dim1 hi16(0) | tile_dim0 (=256)
  g1[4] = ROWS;                                         // tile_dim1 = 128 rows
  g1[5] = G;                                            // tensor_dim0_stride lo32 (=256)
  g1[6] = 0;
  g1[7] = 0;
  i32x4 z4 = {0, 0, 0, 0};
#if __clang_major__ >= 23
  i32x8 z8 = {0, 0, 0, 0, 0, 0, 0, 0};
  __builtin_amdgcn_tensor_load_to_lds(g0, g1, z4, z4, z8, 0);
#else
  __builtin_amdgcn_tensor_load_to_lds(g0, g1, z4, z4, 0);
#endif
}

// ---------- 1) init scratch ----------
__global__ void k_init(int* mkey, float* ssum, float* acc) {
  int t = blockIdx.x * blockDim.x + threadIdx.x;
  if (t < NCELL) { mkey[t] = (int)0x80000000; ssum[t] = 0.f; }
  if (t < 2) acc[t] = 0.f;
}

// ---------- 2) build WMMA-fragment-swizzled B panels ----------
// Panel layout: element for (chunk kc, lane l, j) at [kc*64 + l*2 + j],
// holding B[g = kc*4 + (l>>4)*2 + j, col = l&15]. One b64/lane/frag in the GEMM.
__global__ void k_precompute(const float* __restrict__ Mu, const float* __restrict__ Var,
                             const float* __restrict__ Wl, const float* __restrict__ Wr,
                             float* __restrict__ bsw) {
  int t = blockIdx.x * blockDim.x + threadIdx.x;
  if (t >= PANEL) return;
  int kc = t >> 6, l = (t >> 1) & 31, j = t & 1;
  int g  = kc * 4 + ((l >> 4) * 2) + j;
  int c  = l & 15;
  size_t idx = (size_t)kc * 64 + l * 2 + j;
  // panel 0: Wl||Wr (cols 0..7 = lin_l, 8..15 = lin_r)
  bsw[idx] = (c < H) ? Wl[g * H + c] : Wr[g * H + (c - H)];
  // panels 1/2: invVar^T col tiles {0..15},{16..31};  3/4: (Mu/Var)^T tiles
  float var0 = Var[c * G + g];
  float iv0  = 1.f / var0;
  bsw[1 * PANEL + idx] = iv0;
  bsw[3 * PANEL + idx] = Mu[c * G + g] * iv0;
  int c1 = 16 + c;
  float iv1 = 0.f, mv1 = 0.f;
  if (c1 < K) {
    iv1 = 1.f / Var[c1 * G + g];
    mv1 = Mu[c1 * G + g] * iv1;
  }
  bsw[2 * PANEL + idx] = iv1;
  bsw[4 * PANEL + idx] = mv1;
}

// ---------- 3) per-class constant C[k] = sum_g Mu^2/Var ----------
__global__ void k_classconst(const float* __restrict__ Mu, const float* __restrict__ Var,
                             float* Cc) {
  int k = blockIdx.x;                            // <<<K, G>>>
  float m = Mu[k * G + threadIdx.x];
  float v = m * m / Var[k * G + threadIdx.x];
  v = block_reduce_add(v);
  if (threadIdx.x == 0) Cc[k] = v;
}

// ---------- 4) row softmax: P and log(P+1e-8) ----------
__global__ void k_softmax(const float* __restrict__ W, float* __restrict__ P,
                          float* __restrict__ logP) {
  int n = blockIdx.x * blockDim.x + threadIdx.x;
  if (n >= NCELL) return;
  const float* w = W + (size_t)n * K;
  float m = -3.402823466e+38f;
  #pragma unroll
  for (int k = 0; k < K; ++k) m = fmaxf(m, w[k]);
  float ex[K]; float s = 0.f;
  #pragma unroll
  for (int k = 0; k < K; ++k) { ex[k] = __expf(w[k] - m); s += ex[k]; }
  float inv = 1.f / s;
  #pragma unroll
  for (int k = 0; k < K; ++k) {
    float p = ex[k] * inv;
    P[(size_t)n * K + k] = p;
    logP[(size_t)n * K + k] = __logf(p + 1e-8f);
  }
}

// ---------- 5) fused node GEMMs: TDM-stage X tile, 5 WMMA accumulators ----------
// Per 128-row block: X read ONCE; computes xl||xr, X^2*invVar^T (2 tiles),
// X*(Mu/Var)^T (2 tiles); folds ll_prot partial sum.
__global__ void __launch_bounds__(256) k_node(
    const float* __restrict__ X, const float* __restrict__ S,
    const float* __restrict__ P, const float* __restrict__ bsw,
    const float* __restrict__ Cc, const float* __restrict__ bl,
    const float* __restrict__ br, float* __restrict__ xlr, float* ll_acc) {
  __shared__ float ldsX[ROWS * LDSPITCH];       // 128 x 260 floats (16B row pad)
  int lane = threadIdx.x & 31;
  int wv   = threadIdx.x >> 5;
  int tile = blockIdx.x * TPB + wv;
  int blockRow = blockIdx.x * ROWS;
  bool active = (tile < NT);                    // wave-uniform

  if (threadIdx.x < 32) {                       // wave 0 drives the DMA
    int rem = NCELL - blockRow;
    if (rem > ROWS) rem = ROWS;                 // OOB rows read back as zero
    unsigned ldsOff = (unsigned)(unsigned long long)(void*)ldsX;
    tdm_load_rows(X + (size_t)blockRow * G, ldsOff, rem);
    __builtin_amdgcn_s_wait_tensorcnt(0);
  }
  __syncthreads();                              // tile visible to all 8 waves

  float part = 0.f;
  if (active) {
    int mlo = lane & 15;
    int kk  = (lane >> 4) * 2;                  // A K-slots {0,1} or {2,3}
    const float* arow = ldsX + (size_t)(wv * 16 + mlo) * LDSPITCH + kk;
    const float* bp   = bsw + lane * 2;         // swizzled: one b64 per fragment
    v8f aw = {}, a0 = {}, a1 = {}, b0 = {}, b1 = {};
    for (int kc = 0; kc < CH; ++kc) {
      v2f ax  = *(const v2f*)(arow + kc * 4);   // ds_load_b64 from staged tile
      v2f ax2 = ax * ax;
      const float* bc = bp + kc * 64;
      v2f bw  = *(const v2f*)(bc);
      v2f bv0 = *(const v2f*)(bc + 1 * PANEL);
      v2f bv1 = *(const v2f*)(bc + 2 * PANEL);
      v2f bm0 = *(const v2f*)(bc + 3 * PANEL);
      v2f bm1 = *(const v2f*)(bc + 4 * PANEL);
      aw = __builtin_amdgcn_wmma_f32_16x16x4_f32(false, ax,  false, bw,  (short)0, aw, false, false);
      a0 = __builtin_amdgcn_wmma_f32_16x16x4_f32(false, ax2, false, bv0, (short)0, a0, false, false);
      a1 = __builtin_amdgcn_wmma_f32_16x16x4_f32(false, ax2, false, bv1, (short)0, a1, false, false);
      b0 = __builtin_amdgcn_wmma_f32_16x16x4_f32(false, ax,  false, bm0, (short)0, b0, false, false);
      b1 = __builtin_amdgcn_wmma_f32_16x16x4_f32(false, ax,  false, bm1, (short)0, b1, false, false);
    }
    // D layout: lanes 0-15 hold M=r, lanes 16-31 hold M=8+r; N = lane&15
    int ncol  = mlo;
    int mbase = (lane >> 4) * 8;
    float bias = (ncol < H) ? bl[ncol] : br[ncol - H];
    #pragma unroll
    for (int r = 0; r < 8; ++r) {
      int n = tile * 16 + mbase + r;
      xlr[(size_t)n * 16 + ncol] = aw[r] + bias;
      float sn = S[n];
      {                                          // class cols 0..15: always < K
        float F = -0.5f * (a0[r] - 2.f * sn * b0[r] + sn * sn * Cc[ncol]);
        part += P[(size_t)n * K + ncol] * F;
      }
      int c1 = 16 + ncol;                        // class cols 16..31: valid iff < K
      if (c1 < K) {
        float F = -0.5f * (a1[r] - 2.f * sn * b1[r] + sn * sn * Cc[c1]);
        part += P[(size_t)n * K + c1] * F;
      }
    }
  }
  part = block_reduce_add(part);                 // all waves participate (barrier)
  if (threadIdx.x == 0) atomicAdd(ll_acc, part);
}

// ---------- 6) edge scores + segment max ----------
__global__ void k_edge_e(const int* __restrict__ EI, const float* __restrict__ xlr,
                         const float* __restrict__ att, float* __restrict__ eex,
                         int* __restrict__ mkey) {
  int e = blockIdx.x * blockDim.x + threadIdx.x;
  if (e >= E) return;
  int r = EI[e], c = EI[E + e];
  const float* xl = xlr + (size_t)r * 16;        // cols 0..7 = lin_l(source)
  const float* xr = xlr + (size_t)c * 16 + H;    // cols 8..15 = lin_r(target)
  float s = 0.f;
  #pragma unroll
  for (int h = 0; h < H; ++h) {
    float v = xl[h] + xr[h];
    v = (v > 0.f) ? v : NEG_SLOPE * v;
    s += v * att[h];
  }
  eex[e] = s;
  atomicMax(&mkey[c], fkey(s));
}

// ---------- 7) exp + segment sum ----------
__global__ void k_edge_exp(const int* __restrict__ EI, float* __restrict__ eex,
                           const int* __restrict__ mkey, float* __restrict__ ssum) {
  int e = blockIdx.x * blockDim.x + threadIdx.x;
  if (e >= E) return;
  int c = EI[E + e];
  float ex = __expf(eex[e] - fdec(mkey[c]));
  eex[e] = ex;
  atomicAdd(&ssum[c], ex);
}

// ---------- 8) edge cross-entropy ----------
__global__ void k_edge_ce(const int* __restrict__ EI, const float* __restrict__ eex,
                          const float* __restrict__ ssum, const float* __restrict__ P,
                          const float* __restrict__ logP, float* ce_acc) {
  int e = blockIdx.x * blockDim.x + threadIdx.x;
  float part = 0.f;
  if (e < E) {
    int r = EI[e], c = EI[E + e];
    float alpha = eex[e] / (ssum[c] + 1e-16f);
    const float* pr = P + (size_t)r * K;
    const float* lc = logP + (size_t)c * K;
    float dot = 0.f;
    #pragma unroll
    for (int k = 0; k < K; ++k) dot += pr[k] * lc[k];
    part = alpha * dot;
  }
  part = block_reduce_add(part);
  if (threadIdx.x == 0) atomicAdd(ce_acc, part);
}

// ---------- 9) finalize scalars ----------
__global__ void k_final(const float* __restrict__ acc, float* __restrict__ out) {
  if (threadIdx.x == 0) {
    out[0] = acc[0] / (float)NCELL;       // ll_prot
    out[1] = -acc[1] / (float)NCELL;      // ce_space
  }
}

extern "C" void kernel_launch(void* const* d_in, const int* in_sizes, int n_in,
                              void* d_out, int out_size, void* d_ws, size_t ws_size,
                              hipStream_t stream) {
  const float* X   = (const float*)d_in[0];
  const float* Mu  = (const float*)d_in[1];
  const float* Var = (const float*)d_in[2];
  const int*   EI  = (const int*)d_in[3];
  const float* W   = (const float*)d_in[4];
  const float* S   = (const float*)d_in[5];
  const float* Wl  = (const float*)d_in[6];
  const float* bl  = (const float*)d_in[7];
  const float* Wr  = (const float*)d_in[8];
  const float* br  = (const float*)d_in[9];
  const float* att = (const float*)d_in[10];

  float* out = (float*)d_out;
  float* ws  = (float*)d_ws;

  float* bsw  = ws + OFF_BSW;
  float* Cc   = ws + OFF_CC;
  float* logP = ws + OFF_LOGP;
  float* xlr  = ws + OFF_XLR;
  float* eex  = ws + OFF_EEX;
  int*   mkey = (int*)(ws + OFF_MKEY);
  float* ssum = ws + OFF_SSUM;
  float* acc  = ws + OFF_ACC;
  float* P    = out + 2;

  const int TB = 256;
  k_init      <<<(NCELL + TB - 1) / TB, TB, 0, stream>>>(mkey, ssum, acc);
  k_precompute<<<(PANEL + TB - 1) / TB, TB, 0, stream>>>(Mu, Var, Wl, Wr, bsw);
  k_classconst<<<K, G, 0, stream>>>(Mu, Var, Cc);
  k_softmax   <<<(NCELL + TB - 1) / TB, TB, 0, stream>>>(W, P, logP);

  int nodeBlocks = (NT + TPB - 1) / TPB;           // 157
  k_node <<<nodeBlocks, TB, 0, stream>>>(X, S, P, bsw, Cc, bl, br, xlr, acc);

  k_edge_e  <<<(E + TB - 1) / TB, TB, 0, stream>>>(EI, xlr, att, eex, mkey);
  k_edge_exp<<<(E + TB - 1) / TB, TB, 0, stream>>>(EI, eex, mkey, ssum);
  k_edge_ce <<<(E + TB - 1) / TB, TB, 0, stream>>>(EI, eex, ssum, P, logP, acc + 1);
  k_final   <<<1, 32, 0, stream>>>(acc, out);
}